// Yolo3_62947040690195
// MI455X (gfx1250) — compile-verified
//
#include <hip/hip_runtime.h>
#include <hip/hip_bf16.h>
#include <math.h>

// ---------------------------------------------------------------------------
// YOLOv3 decode + top-k + NMS for MI455X (gfx1250, wave32, 320KB LDS/WGP)
//   input : (32, 255, 64, 64) f32
//   output: dets (32,1024,7) f32  ++  keep (32,1024) f32   (flat concat)
// ---------------------------------------------------------------------------

#define BATCH   32
#define GSZ     64
#define NANC    (GSZ*GSZ*3)          // 12288 anchors / image (= 12 * 1024)
#define TOPK    1024
#define NCLS    80
#define STRIDEF 8.0f                 // 512 / 64
#define CLS_OFF 10000.0f
#define IOU_T   0.4f

// CDNA5 async global->LDS path (ASYNCcnt), guarded so either toolchain compiles.
#if defined(__AMDGCN__) && __has_builtin(__builtin_amdgcn_global_load_async_to_lds_b128)
#define USE_ASYNC_LDS 1
typedef int v4i_async __attribute__((vector_size(16)));
typedef __attribute__((address_space(1))) v4i_async* gv4p_t;
typedef __attribute__((address_space(3))) v4i_async* lv4p_t;
#else
#define USE_ASYNC_LDS 0
#endif

#if defined(__AMDGCN__) && __has_builtin(__builtin_amdgcn_global_load_async_to_lds_b64)
#define USE_ASYNC_LDS64 1
typedef int v2i_async __attribute__((vector_size(8)));
typedef __attribute__((address_space(1))) v2i_async* gv2p_t;
typedef __attribute__((address_space(3))) v2i_async* lv2p_t;
#else
#define USE_ASYNC_LDS64 0
#endif

__device__ __forceinline__ void wait_async0() {
#if defined(__AMDGCN__)
#if __has_builtin(__builtin_amdgcn_s_wait_asynccnt)
  __builtin_amdgcn_s_wait_asynccnt(0);
#else
  asm volatile("s_wait_asynccnt 0" ::: "memory");
#endif
#endif
}

// ---------------------------------------------------------------------------
// Kernel 1: fused decode. One thread per (anchor a, g1, g2); consecutive lanes
// hit consecutive g2 -> every channel read is fully coalesced (134MB, one pass).
// ---------------------------------------------------------------------------
__global__ __launch_bounds__(256) void yolo_decode(
    const float* __restrict__ x,
    float* __restrict__ det,                 // (B, NANC, 8)
    unsigned long long* __restrict__ keys)   // (B, NANC)
{
  const int b    = blockIdx.y;
  const int flat = blockIdx.x * 256 + threadIdx.x;   // [0, 12288)
  const int a    = flat >> 12;                       // anchor 0..2
  const int pos  = flat & 4095;                      // g1*64 + g2
  const int g1   = pos >> 6;
  const int g2   = pos & 63;

  const float* base = x + ((size_t)b * 255 + (size_t)a * 85) * 4096 + pos;

  const float tx = base[0];
  const float ty = base[4096];
  const float tw = base[2 * 4096];
  const float th = base[3 * 4096];
  const float cl = base[4 * 4096];

  const float conf = 1.0f / (1.0f + __expf(-cl));

  // reference row order: n = (g2*G + g1)*3 + a, offsets from tiled meshgrid
  const int n = ((g2 << 6) + g1) * 3 + a;
  const float xo = (float)((n >> 6) & 63);
  const float yo = (float)(n & 63);

  const float cx = tx + xo;
  const float cy = ty + yo;
  const float hw = 0.5f * tw;
  const float hh = 0.5f * th;

  float maxv = -INFINITY;
  int   maxi = 0;
#pragma unroll 8
  for (int d = 0; d < NCLS; ++d) {
    const float v = base[(size_t)(5 + d) * 4096];
    if (v > maxv) { maxv = v; maxi = d; }       // first-max wins (argmax tie rule)
  }

  float4* dr = (float4*)(det + ((size_t)b * NANC + n) * 8);
  dr[0] = make_float4(cx - hw, cy - hh, cx + hw, cy + hh);
  dr[1] = make_float4(conf, maxv, (float)maxi, 0.0f);

  // Sort key: descending (score, then lower index). Monotonic float encode;
  // low 32 bits = ~n makes every key unique -> exact radix-select boundary.
  const float score = (conf > 0.5f) ? conf : -INFINITY;
  unsigned u = __float_as_uint(score);
  u = ((int)u >= 0) ? (u | 0x80000000u) : ~u;
  keys[(size_t)b * NANC + n] =
      ((unsigned long long)u << 32) | (unsigned)(~n);
}

// ---------------------------------------------------------------------------
// Kernel 2: per-image exact top-1024 via MSB-first radix-select over LDS-
// resident keys (96 KB), then a 1024-element bitonic sort of the winners.
// ~20x less LDS traffic than full 16K bitonic sort (the old serial hotspot).
// ---------------------------------------------------------------------------
#define K2_KEYS   0                         // 12288 * 8  = 98304
#define K2_TOP    98304                     // 1024 * 8   = 8192   -> 106496
#define K2_BINS   106496                    // 256 * 4    = 1024   -> 107520
#define K2_PFX    107520                    // u64        = 8      -> 107528
#define K2_REM    107528                    // u32        = 4
#define K2_CNT    107532                    // u32        = 4
#define K2_SMEM   107552

__global__ __launch_bounds__(1024) void yolo_topk(
    const unsigned long long* __restrict__ keys,
    const float* __restrict__ det,
    float* __restrict__ cand)                 // (B, TOPK, 8); slot7 = valid
{
  extern __shared__ unsigned char smem2[];
  unsigned long long* s_keys = (unsigned long long*)(smem2 + K2_KEYS);
  unsigned long long* s_top  = (unsigned long long*)(smem2 + K2_TOP);
  unsigned*           s_bins = (unsigned*)(smem2 + K2_BINS);
  unsigned long long* s_pfx  = (unsigned long long*)(smem2 + K2_PFX);
  unsigned*           s_rem  = (unsigned*)(smem2 + K2_REM);
  unsigned*           s_cnt  = (unsigned*)(smem2 + K2_CNT);

  const int b   = blockIdx.x;
  const int tid = threadIdx.x;
  const unsigned long long* kb = keys + (size_t)b * NANC;

  // stage all 12288 keys into LDS (async b64 path)
#pragma unroll
  for (int p = 0; p < 12; ++p) {
    const int e = tid + p * 1024;
#if USE_ASYNC_LDS64
    __builtin_amdgcn_global_load_async_to_lds_b64(
        (gv2p_t)(void*)(kb + e), (lv2p_t)(void*)(s_keys + e), 0, 0);
#else
    s_keys[e] = kb[e];
#endif
  }
#if USE_ASYNC_LDS64
  wait_async0();
#endif
  if (tid == 0) { *s_pfx = 0ull; *s_rem = TOPK; }
  __syncthreads();

  // 8 MSB-first passes of 256-bin histogram -> exact 1024th-largest key T
  for (int d = 7; d >= 0; --d) {
    if (tid < 256) s_bins[tid] = 0u;
    __syncthreads();
    const unsigned long long pfx = *s_pfx;
    const int shift = d * 8;
#pragma unroll
    for (int p = 0; p < 12; ++p) {
      const unsigned long long k = s_keys[tid + p * 1024];
      const bool match = (d == 7) || ((k >> (shift + 8)) == pfx);
      if (match)
        atomicAdd(&s_bins[(unsigned)((k >> shift) & 255ull)], 1u);
    }
    __syncthreads();
    if (tid == 0) {
      const unsigned rem0 = *s_rem;
      unsigned acc = 0, g = 0;
      for (int gg = 255; gg >= 0; --gg) {        // walk digits high -> low
        const unsigned c = s_bins[gg];
        if (acc + c >= rem0) { g = (unsigned)gg; break; }
        acc += c;
      }
      *s_rem = rem0 - acc;                       // rank within chosen digit
      *s_pfx = (pfx << 8) | g;
    }
    __syncthreads();
  }

  const unsigned long long T = *s_pfx;           // keys unique => exactly 1024 with k >= T
  if (tid == 0) *s_cnt = 0u;
  __syncthreads();

  // compact the winners (unsorted), then sort just 1024 of them
#pragma unroll
  for (int p = 0; p < 12; ++p) {
    const unsigned long long k = s_keys[tid + p * 1024];
    if (k >= T) {
      const unsigned posn = atomicAdd(s_cnt, 1u);
      s_top[posn & (TOPK - 1)] = k;
    }
  }
  __syncthreads();

  for (unsigned k = 2; k <= TOPK; k <<= 1) {
    for (unsigned j = k >> 1; j > 0; j >>= 1) {
      const unsigned i = (unsigned)tid;
      const unsigned ixj = i ^ j;
      if (ixj > i) {
        const unsigned long long A = s_top[i];
        const unsigned long long B = s_top[ixj];
        const bool sw = ((i & k) == 0) ? (A < B) : (A > B);  // descending
        if (sw) { s_top[i] = B; s_top[ixj] = A; }
      }
      __syncthreads();
    }
  }

  // rank tid: gather det row, stamp validity into slot 7
  const unsigned long long kk = s_top[tid];
  const unsigned idx = ~(unsigned)kk;
  const unsigned enc = (unsigned)(kk >> 32);
  const float valid = (enc > 0xBF000000u) ? 1.0f : 0.0f;   // enc(0.5f)

  const float4* src = (const float4*)(det + ((size_t)b * NANC + idx) * 8);
  float4 r0 = src[0];
  float4 r1 = src[1];
  r1.w = valid;
  float4* dst = (float4*)(cand + ((size_t)b * TOPK + tid) * 8);
  dst[0] = r0; dst[1] = r1;
}

// ---------------------------------------------------------------------------
// Kernel 3: NMS. Async-stage candidate rows to LDS, build the full 1024x1024
// suppression bit-matrix in LDS (128 KB), then a single wave32 does the greedy
// scan over a 1024-bit bitset (lane w owns word w) — no barriers in the scan.
// ---------------------------------------------------------------------------
#define SM_RAW    0                      // 1024*8 f32  = 32768
#define SM_MASK   32768                  // 1024*32 u32 = 131072
#define SM_BOX    163840                 // 5*1024 f32  = 20480
#define SM_VALID  184320                 // 32 u32      = 128
#define SM_KEEP   184448                 // 32 u32      = 128
#define NMS_SMEM  184576

__global__ __launch_bounds__(1024) void yolo_nms(
    const float* __restrict__ cand,
    float* __restrict__ out)
{
  extern __shared__ unsigned char smem[];
  float*    s_raw   = (float*)(smem + SM_RAW);
  unsigned* s_mask  = (unsigned*)(smem + SM_MASK);
  float*    s_x1    = (float*)(smem + SM_BOX);
  float*    s_y1    = s_x1 + 1024;
  float*    s_x2    = s_y1 + 1024;
  float*    s_y2    = s_x2 + 1024;
  float*    s_ar    = s_y2 + 1024;
  unsigned* s_validm = (unsigned*)(smem + SM_VALID);
  unsigned* s_keep   = (unsigned*)(smem + SM_KEEP);

  const int b   = blockIdx.x;
  const int tid = threadIdx.x;

  const float* src = cand + ((size_t)b * TOPK + tid) * 8;
  float* dstl = s_raw + (size_t)tid * 8;

#if USE_ASYNC_LDS
  __builtin_amdgcn_global_load_async_to_lds_b128(
      (gv4p_t)(void*)src, (lv4p_t)(void*)dstl, 0, 0);
  __builtin_amdgcn_global_load_async_to_lds_b128(
      (gv4p_t)(void*)(src + 4), (lv4p_t)(void*)(dstl + 4), 0, 0);
  wait_async0();
#else
  ((float4*)dstl)[0] = ((const float4*)src)[0];
  ((float4*)dstl)[1] = ((const float4*)src)[1];
#endif
  __syncthreads();

  // class-shifted boxes + areas + validity bitmask (wave32 ballot)
  {
    const float* r = s_raw + (size_t)tid * 8;
    const float off = r[6] * CLS_OFF;
    const float x1 = r[0] + off, y1 = r[1] + off;
    const float x2 = r[2] + off, y2 = r[3] + off;
    s_x1[tid] = x1; s_y1[tid] = y1; s_x2[tid] = x2; s_y2[tid] = y2;
    s_ar[tid] = (x2 - x1) * (y2 - y1);
    const bool valid = (r[7] != 0.0f);
    const unsigned long long bal = __ballot(valid);
    if ((tid & 31) == 0) s_validm[tid >> 5] = (unsigned)bal;
  }
  __syncthreads();

  // row tid of the over-threshold bit-matrix (broadcast LDS reads over j)
  {
    const float ax1 = s_x1[tid], ay1 = s_y1[tid];
    const float ax2 = s_x2[tid], ay2 = s_y2[tid];
    const float aar = s_ar[tid];
    for (int w = 0; w < 32; ++w) {
      unsigned bits = 0u;
      const int jb = w << 5;
#pragma unroll
      for (int t = 0; t < 32; ++t) {
        const int j = jb + t;
        const float lx = fmaxf(ax1, s_x1[j]);
        const float ly = fmaxf(ay1, s_y1[j]);
        const float rx = fminf(ax2, s_x2[j]);
        const float ry = fminf(ay2, s_y2[j]);
        const float iw = fmaxf(rx - lx, 0.0f);
        const float ih = fmaxf(ry - ly, 0.0f);
        const float inter = iw * ih;
        const float iou = inter / (aar + s_ar[j] - inter + 1e-9f);
        if ((iou > IOU_T) && (j > tid)) bits |= (1u << t);
      }
      s_mask[(size_t)tid * 32 + w] = bits;
    }
  }
  __syncthreads();

  // greedy scan: one wave32, lane w owns suppressed word w; conflict-free
  // LDS row fetch + OR per iteration; 'kept' is lane-uniform so no divergence.
  if (tid < 32) {
    unsigned sup = 0u;
    unsigned keepw = 0u;
    for (int i = 0; i < TOPK; ++i) {
      const int wi = i >> 5;
      const int bi = i & 31;
      const unsigned supw = __shfl(sup, wi, 32);
      const unsigned valw = s_validm[wi];
      const bool kept = (((valw >> bi) & 1u) != 0u) &&
                        (((supw >> bi) & 1u) == 0u);
      if (kept) sup |= s_mask[(size_t)i * 32 + tid];
      if (kept && tid == wi) keepw |= (1u << bi);
    }
    s_keep[tid] = keepw;
  }
  __syncthreads();

  // dets = keep ? stride*cand : 0 ;  keep as 0/1 float
  const unsigned kept = (s_keep[tid >> 5] >> (tid & 31)) & 1u;
  const float sc = kept ? STRIDEF : 0.0f;
  const float* r = s_raw + (size_t)tid * 8;
  float* dr = out + ((size_t)b * TOPK + tid) * 7;
#pragma unroll
  for (int c = 0; c < 7; ++c) dr[c] = sc * r[c];
  out[(size_t)BATCH * TOPK * 7 + (size_t)b * TOPK + tid] = kept ? 1.0f : 0.0f;
}

// ---------------------------------------------------------------------------
// Host launcher
// ---------------------------------------------------------------------------
extern "C" void kernel_launch(void* const* d_in, const int* in_sizes, int n_in,
                              void* d_out, int out_size, void* d_ws, size_t ws_size,
                              hipStream_t stream)
{
  (void)in_sizes; (void)n_in; (void)out_size; (void)ws_size;

  const float* x = (const float*)d_in[0];
  float* out = (float*)d_out;
  char* ws = (char*)d_ws;

  // workspace layout (~16 MB total):
  //   det  : B*NANC*8 f32   = 12,582,912 B
  //   keys : B*NANC   u64   =  3,145,728 B
  //   cand : B*TOPK*8 f32   =  1,048,576 B
  float* det = (float*)ws;
  unsigned long long* keys =
      (unsigned long long*)(ws + (size_t)BATCH * NANC * 8 * sizeof(float));
  float* cand = (float*)(ws + (size_t)BATCH * NANC * 8 * sizeof(float)
                            + (size_t)BATCH * NANC * sizeof(unsigned long long));

  // opt in to large dynamic LDS (CDNA5: up to 320 KB per workgroup)
  (void)hipFuncSetAttribute((const void*)yolo_topk,
                            hipFuncAttributeMaxDynamicSharedMemorySize,
                            K2_SMEM);
  (void)hipFuncSetAttribute((const void*)yolo_nms,
                            hipFuncAttributeMaxDynamicSharedMemorySize,
                            NMS_SMEM);

  dim3 gdec(NANC / 256, BATCH);
  yolo_decode<<<gdec, 256, 0, stream>>>(x, det, keys);
  yolo_topk<<<BATCH, 1024, K2_SMEM, stream>>>(keys, det, cand);
  yolo_nms<<<BATCH, 1024, NMS_SMEM, stream>>>(cand, out);
}